// LocalGlobalRegistration_56341380989503
// MI455X (gfx1250) — compile-verified
//
#include <hip/hip_runtime.h>
#include <math.h>

// ---------------------------------------------------------------------------
// LocalGlobalRegistration for MI455X (gfx1250), B=128 patches, M=N=64, 3-D.
// Pairwise sq-distance tiles use V_WMMA_F32_16X16X4_F32 (K=4: xyz + 0 pad),
// with operands built branchlessly in registers (no exec-masked LDS loads).
// All reductions are fixed-order (deterministic); no float atomics anywhere.
// ---------------------------------------------------------------------------

#define BB 128
#define MM 64
#define NN 64
#define R2ACC 0.01f       // ACCEPTANCE_RADIUS^2
#define CONF_TH 0.05f
#define EPSV 1e-5f

typedef __attribute__((ext_vector_type(2))) float v2f;
typedef __attribute__((ext_vector_type(8))) float v8f;

// A (16x4 f32): lanes 0-15 -> {K0,K1}, lanes 16-31 -> {K2,K3}; B mirrored.
__device__ __forceinline__ v8f wmma_tile(v2f a, v2f b) {
  v8f c = {};
  // (neg_a, A, neg_b, B, c_mod, C, reuse_a, reuse_b)
  return __builtin_amdgcn_wmma_f32_16x16x4_f32(false, a, false, b,
                                               (short)0, c, false, false);
}

// ------------------------------ 3x3 SVD ------------------------------------
__device__ __forceinline__ float det3(const float V[3][3]) {
  return V[0][0] * (V[1][1] * V[2][2] - V[1][2] * V[2][1])
       - V[0][1] * (V[1][0] * V[2][2] - V[1][2] * V[2][0])
       + V[0][2] * (V[1][0] * V[2][1] - V[1][1] * V[2][0]);
}

__device__ void jacobi_rot(float A[3][3], float V[3][3], int p, int q) {
  float apq = A[p][q];
  if (fabsf(apq) < 1e-30f) return;
  float tau = (A[q][q] - A[p][p]) / (2.0f * apq);
  float root = sqrtf(1.0f + tau * tau);
  float t = (tau >= 0.0f) ? 1.0f / (tau + root) : 1.0f / (tau - root);
  float c = rsqrtf(1.0f + t * t);
  float s = t * c;
  for (int k = 0; k < 3; ++k) {
    float akp = A[k][p], akq = A[k][q];
    A[k][p] = c * akp - s * akq;
    A[k][q] = s * akp + c * akq;
  }
  for (int k = 0; k < 3; ++k) {
    float apk = A[p][k], aqk = A[q][k];
    A[p][k] = c * apk - s * aqk;
    A[q][k] = s * apk + c * aqk;
  }
  for (int k = 0; k < 3; ++k) {
    float vkp = V[k][p], vkq = V[k][q];
    V[k][p] = c * vkp - s * vkq;
    V[k][q] = s * vkp + c * vkq;
  }
}

// R = V' E U^T per torch/JAX convention (H = U S V^T), t = cr - R cs.
// det(V)=+1 enforced; U built with u2=cross(u0,u1) so det(U)=+1; this
// reproduces the reference's diag(1,1,sign(det(V U^T))) in all sign cases.
__device__ void svd_rigid_rt(const float Hin[9], const float cr[3],
                             const float cs[3], float Rout[9], float tout[3]) {
  float H[3][3] = {{Hin[0], Hin[1], Hin[2]},
                   {Hin[3], Hin[4], Hin[5]},
                   {Hin[6], Hin[7], Hin[8]}};
  float A[3][3], V[3][3];
  for (int i = 0; i < 3; ++i)
    for (int j = 0; j < 3; ++j) {
      float s = 0.0f;
      for (int k = 0; k < 3; ++k) s += H[k][i] * H[k][j];
      A[i][j] = s;
      V[i][j] = (i == j) ? 1.0f : 0.0f;
    }
  for (int sw = 0; sw < 8; ++sw) {
    jacobi_rot(A, V, 0, 1);
    jacobi_rot(A, V, 0, 2);
    jacobi_rot(A, V, 1, 2);
  }
  float lam[3] = {A[0][0], A[1][1], A[2][2]};
  for (int i = 0; i < 2; ++i)
    for (int j = 0; j < 2 - i; ++j)
      if (lam[j] < lam[j + 1]) {
        float tl = lam[j]; lam[j] = lam[j + 1]; lam[j + 1] = tl;
        for (int k = 0; k < 3; ++k) {
          float tv = V[k][j]; V[k][j] = V[k][j + 1]; V[k][j + 1] = tv;
        }
      }
  if (det3(V) < 0.0f) {
    V[0][2] = -V[0][2]; V[1][2] = -V[1][2]; V[2][2] = -V[2][2];
  }
  float b0[3], b1[3], u0[3], u1[3], u2[3];
  for (int c = 0; c < 3; ++c) {
    b0[c] = H[c][0] * V[0][0] + H[c][1] * V[1][0] + H[c][2] * V[2][0];
    b1[c] = H[c][0] * V[0][1] + H[c][1] * V[1][1] + H[c][2] * V[2][1];
  }
  float n0 = sqrtf(b0[0] * b0[0] + b0[1] * b0[1] + b0[2] * b0[2]);
  if (n0 > 1e-12f) {
    float inv = 1.0f / n0;
    u0[0] = b0[0] * inv; u0[1] = b0[1] * inv; u0[2] = b0[2] * inv;
  } else {
    u0[0] = 1.0f; u0[1] = 0.0f; u0[2] = 0.0f;
  }
  float proj = b1[0] * u0[0] + b1[1] * u0[1] + b1[2] * u0[2];
  b1[0] -= proj * u0[0]; b1[1] -= proj * u0[1]; b1[2] -= proj * u0[2];
  float n1 = sqrtf(b1[0] * b1[0] + b1[1] * b1[1] + b1[2] * b1[2]);
  if (n1 > 1e-12f) {
    float inv = 1.0f / n1;
    u1[0] = b1[0] * inv; u1[1] = b1[1] * inv; u1[2] = b1[2] * inv;
  } else {
    float e[3] = {0.0f, 0.0f, 0.0f};
    e[(fabsf(u0[0]) < 0.9f) ? 0 : 1] = 1.0f;
    float cx = u0[1] * e[2] - u0[2] * e[1];
    float cy = u0[2] * e[0] - u0[0] * e[2];
    float cz = u0[0] * e[1] - u0[1] * e[0];
    float nn = rsqrtf(cx * cx + cy * cy + cz * cz + 1e-30f);
    u1[0] = cx * nn; u1[1] = cy * nn; u1[2] = cz * nn;
  }
  u2[0] = u0[1] * u1[2] - u0[2] * u1[1];
  u2[1] = u0[2] * u1[0] - u0[0] * u1[2];
  u2[2] = u0[0] * u1[1] - u0[1] * u1[0];
  for (int i = 0; i < 3; ++i)
    for (int j = 0; j < 3; ++j)
      Rout[i * 3 + j] = V[i][0] * u0[j] + V[i][1] * u1[j] + V[i][2] * u2[j];
  for (int i = 0; i < 3; ++i)
    tout[i] = cr[i] - (Rout[i * 3 + 0] * cs[0] + Rout[i * 3 + 1] * cs[1] +
                       Rout[i * 3 + 2] * cs[2]);
}

__device__ __forceinline__ void set_bit64(unsigned& lo, unsigned& hi, int i) {
  if (i < 32) lo |= 1u << i; else hi |= 1u << (i - 32);
}

// ---------------- K1: exp + mutual top-3 + confidence -> w -----------------
__global__ void lgr_corr_kernel(const float* __restrict__ score,
                                float* __restrict__ w,
                                int* __restrict__ corrcnt) {
  int b = blockIdx.x, t = threadIdx.x;  // 64 threads
  __shared__ unsigned refA[64], refB[64], srcA[64], srcB[64];
  __shared__ int red[64];
  const float* sb = score + b * MM * NN;

  float v0 = -1e30f, v1 = -1e30f, v2 = -1e30f;
  int i0 = 0, i1 = 0, i2 = 0;
  for (int n = 0; n < NN; ++n) {
    float e = expf(sb[t * NN + n]);
    if (e > v0)      { v2 = v1; i2 = i1; v1 = v0; i1 = i0; v0 = e; i0 = n; }
    else if (e > v1) { v2 = v1; i2 = i1; v1 = e; i1 = n; }
    else if (e > v2) { v2 = e; i2 = n; }
  }
  unsigned lo = 0, hi = 0;
  set_bit64(lo, hi, i0); set_bit64(lo, hi, i1); set_bit64(lo, hi, i2);
  refA[t] = lo; refB[t] = hi;

  v0 = v1 = v2 = -1e30f; i0 = i1 = i2 = 0;
  for (int m = 0; m < MM; ++m) {
    float e = expf(sb[m * NN + t]);
    if (e > v0)      { v2 = v1; i2 = i1; v1 = v0; i1 = i0; v0 = e; i0 = m; }
    else if (e > v1) { v2 = v1; i2 = i1; v1 = e; i1 = m; }
    else if (e > v2) { v2 = e; i2 = m; }
  }
  lo = 0; hi = 0;
  set_bit64(lo, hi, i0); set_bit64(lo, hi, i1); set_bit64(lo, hi, i2);
  srcA[t] = lo; srcB[t] = hi;
  __syncthreads();

  int cnt = 0;
  float* wb = w + b * MM * NN;
  for (int n = 0; n < NN; ++n) {
    float e = expf(sb[t * NN + n]);
    unsigned rs = (n < 32) ? ((refA[t] >> n) & 1u) : ((refB[t] >> (n - 32)) & 1u);
    unsigned ss = (t < 32) ? ((srcA[n] >> t) & 1u) : ((srcB[n] >> (t - 32)) & 1u);
    unsigned c = rs & ss & (unsigned)(e > CONF_TH);
    wb[t * NN + n] = c ? e : 0.0f;
    cnt += (int)c;
  }
  red[t] = cnt;
  __syncthreads();
  for (int s = 32; s > 0; s >>= 1) {
    if (t < s) red[t] += red[t + s];
    __syncthreads();
  }
  if (t == 0) corrcnt[b] = red[0];
}

// ------------- K2: per-patch weighted Procrustes -> Rb, tb -----------------
__global__ void lgr_procrustes_batch_kernel(const float* __restrict__ ref,
                                            const float* __restrict__ src,
                                            const float* __restrict__ w,
                                            float* __restrict__ Rb,
                                            float* __restrict__ tb) {
  int b = blockIdx.x, t = threadIdx.x;  // 64 threads
  __shared__ float red[64][12];
  const float* wb = w + b * MM * NN;
  const float* refb = ref + b * MM * 3;
  const float* srcb = src + b * NN * 3;

  float rw = 0.0f, cw = 0.0f;
  for (int n = 0; n < NN; ++n) rw += wb[t * NN + n];
  for (int m = 0; m < MM; ++m) cw += wb[m * NN + t];
  red[t][0] = rw;
  red[t][1] = rw * refb[t * 3 + 0];
  red[t][2] = rw * refb[t * 3 + 1];
  red[t][3] = rw * refb[t * 3 + 2];
  red[t][4] = cw * srcb[t * 3 + 0];
  red[t][5] = cw * srcb[t * 3 + 1];
  red[t][6] = cw * srcb[t * 3 + 2];
  red[t][7] = 0.0f;
  __syncthreads();
  for (int s = 32; s > 0; s >>= 1) {
    if (t < s) for (int j = 0; j < 8; ++j) red[t][j] += red[t + s][j];
    __syncthreads();
  }
  float Sp = red[0][0] + EPSV;
  float cr0 = red[0][1] / Sp, cr1 = red[0][2] / Sp, cr2 = red[0][3] / Sp;
  float cs0 = red[0][4] / Sp, cs1 = red[0][5] / Sp, cs2 = red[0][6] / Sp;
  __syncthreads();

  float a0 = 0.0f, a1 = 0.0f, a2 = 0.0f;
  for (int m = 0; m < MM; ++m) {
    float wv = wb[m * NN + t];
    a0 += wv * (refb[m * 3 + 0] - cr0);
    a1 += wv * (refb[m * 3 + 1] - cr1);
    a2 += wv * (refb[m * 3 + 2] - cr2);
  }
  float sc0 = srcb[t * 3 + 0] - cs0;
  float sc1 = srcb[t * 3 + 1] - cs1;
  float sc2 = srcb[t * 3 + 2] - cs2;
  red[t][0] = sc0 * a0; red[t][1] = sc0 * a1; red[t][2] = sc0 * a2;
  red[t][3] = sc1 * a0; red[t][4] = sc1 * a1; red[t][5] = sc1 * a2;
  red[t][6] = sc2 * a0; red[t][7] = sc2 * a1; red[t][8] = sc2 * a2;
  red[t][9] = red[t][10] = red[t][11] = 0.0f;
  __syncthreads();
  for (int s = 32; s > 0; s >>= 1) {
    if (t < s) for (int j = 0; j < 9; ++j) red[t][j] += red[t + s][j];
    __syncthreads();
  }
  if (t == 0) {
    float H[9], cr[3] = {cr0, cr1, cr2}, cs[3] = {cs0, cs1, cs2};
    for (int j = 0; j < 9; ++j) H[j] = red[0][j];
    float R[9], tt[3];
    svd_rigid_rt(H, cr, cs, R, tt);
    for (int j = 0; j < 9; ++j) Rb[b * 9 + j] = R[j];
    for (int j = 0; j < 3; ++j) tb[b * 3 + j] = tt[j];
  }
}

// ---- shared LDS staging for the WMMA distance kernels ---------------------
#define STAGE_POINTS()                                                        \
  for (int i = lane; i < 64; i += 32) {                                       \
    float sxx = src[(b * NN + i) * 3 + 0];                                    \
    float syy = src[(b * NN + i) * 3 + 1];                                    \
    float szz = src[(b * NN + i) * 3 + 2];                                    \
    sx[i] = sxx; sy[i] = syy; sz[i] = szz;                                    \
    float mxx = R00 * sxx + R01 * syy + R02 * szz + t0;                       \
    float myy = R10 * sxx + R11 * syy + R12 * szz + t1;                       \
    float mzz = R20 * sxx + R21 * syy + R22 * szz + t2;                       \
    mx[i] = mxx; my[i] = myy; mz[i] = mzz;                                    \
    s2m[i] = mxx * mxx + myy * myy + mzz * mzz;                               \
    float rxx = ref[(b * MM + i) * 3 + 0];                                    \
    float ryy = ref[(b * MM + i) * 3 + 1];                                    \
    float rzz = ref[(b * MM + i) * 3 + 2];                                    \
    rx[i] = rxx; ry[i] = ryy; rz[i] = rzz;                                    \
    r2r[i] = rxx * rxx + ryy * ryy + rzz * rzz;                               \
  }

// Build A/B WMMA operands branchlessly: unconditional LDS reads + register
// selects (v_cndmask), no exec-masked loads. idx = i*16 + (lane&15).
#define BUILD_OPS()                                                           \
  v2f aop[4], bop[4];                                                         \
  float s2loc[4];                                                             \
  _Pragma("unroll")                                                           \
  for (int i = 0; i < 4; ++i) {                                               \
    int idx = i * 16 + l;                                                     \
    float ax = rx[idx], ay = ry[idx], az = rz[idx];                           \
    aop[i].x = half ? az : ax;                                                \
    aop[i].y = half ? 0.0f : ay;                                              \
    float bx = mx[idx], by = my[idx], bz = mz[idx];                           \
    bop[i].x = half ? bz : bx;                                                \
    bop[i].y = half ? 0.0f : by;                                              \
    s2loc[i] = s2m[idx];                                                      \
  }

// ------ K3: verify candidate p vs patch b (WMMA distance tiles) ------------
__global__ void lgr_verify_kernel(const float* __restrict__ ref,
                                  const float* __restrict__ src,
                                  const float* __restrict__ w,
                                  const float* __restrict__ Rb,
                                  const float* __restrict__ tb,
                                  int* __restrict__ pcount) {
  int b = blockIdx.x, p = blockIdx.y, lane = threadIdx.x;  // 32 threads
  __shared__ float rx[64], ry[64], rz[64], sx[64], sy[64], sz[64];
  __shared__ float mx[64], my[64], mz[64], r2r[64], s2m[64];
  __shared__ int icnt[32];
  const float* wb = w + b * MM * NN;
  __builtin_prefetch(wb + lane * 16, 0, 1);  // global_prefetch_b8

  float R00 = Rb[p * 9 + 0], R01 = Rb[p * 9 + 1], R02 = Rb[p * 9 + 2];
  float R10 = Rb[p * 9 + 3], R11 = Rb[p * 9 + 4], R12 = Rb[p * 9 + 5];
  float R20 = Rb[p * 9 + 6], R21 = Rb[p * 9 + 7], R22 = Rb[p * 9 + 8];
  float t0 = tb[p * 3 + 0], t1 = tb[p * 3 + 1], t2 = tb[p * 3 + 2];

  STAGE_POINTS();
  __syncthreads();

  int half = lane >> 4, l = lane & 15;
  BUILD_OPS();

  int cnt = 0;
#pragma unroll
  for (int mi = 0; mi < 4; ++mi) {
    int mbase = mi * 16 + (half << 3);
    float r2loc[8];
#pragma unroll
    for (int j = 0; j < 8; ++j) r2loc[j] = r2r[mbase + j];
#pragma unroll
    for (int ni = 0; ni < 4; ++ni) {
      v8f acc = wmma_tile(aop[mi], bop[ni]);
      int n = ni * 16 + l;
#pragma unroll
      for (int j = 0; j < 8; ++j) {
        float wv = wb[(mbase + j) * NN + n];            // unconditional load
        float d2 = r2loc[j] + s2loc[ni] - 2.0f * acc[j];
        cnt += ((int)(d2 < R2ACC)) & ((int)(wv > 0.0f));  // branchless
      }
    }
  }
  icnt[lane] = cnt;
  __syncthreads();
  if (lane == 0) {
    int s = 0;
    for (int i = 0; i < 32; ++i) s += icnt[i];
    pcount[p * BB + b] = s;
  }
}

// ---------- K4: argmax over candidates (first-max, like jnp.argmax) --------
__global__ void lgr_select_best_kernel(const int* __restrict__ pcount,
                                       const int* __restrict__ corrcnt,
                                       const float* __restrict__ Rb,
                                       const float* __restrict__ tb,
                                       float* __restrict__ curRT) {
  int t = threadIdx.x;  // 128 threads
  __shared__ int tot[BB];
  __shared__ int bestS;
  int acc = 0;
  for (int b = 0; b < BB; ++b) acc += pcount[t * BB + b];
  tot[t] = (corrcnt[t] >= 3) ? acc : -1;
  __syncthreads();
  if (t == 0) {
    int best = 0, bv = tot[0];
    for (int p = 1; p < BB; ++p)
      if (tot[p] > bv) { bv = tot[p]; best = p; }
    bestS = best;
  }
  __syncthreads();
  if (t < 9) curRT[t] = Rb[bestS * 9 + t];
  if (t < 3) curRT[9 + t] = tb[bestS * 3 + t];
}

// ---- K5a: refinement pass 1 (per-patch S, Sr, Ss partials via WMMA) -------
__global__ void lgr_refine_pass1(const float* __restrict__ ref,
                                 const float* __restrict__ src,
                                 const float* __restrict__ w,
                                 const float* __restrict__ curRT,
                                 float* __restrict__ part1) {
  int b = blockIdx.x, lane = threadIdx.x;  // 32 threads
  __shared__ float rx[64], ry[64], rz[64], sx[64], sy[64], sz[64];
  __shared__ float mx[64], my[64], mz[64], r2r[64], s2m[64];
  __shared__ float red[32][8];
  const float* wb = w + b * MM * NN;

  float R00 = curRT[0], R01 = curRT[1], R02 = curRT[2];
  float R10 = curRT[3], R11 = curRT[4], R12 = curRT[5];
  float R20 = curRT[6], R21 = curRT[7], R22 = curRT[8];
  float t0 = curRT[9], t1 = curRT[10], t2 = curRT[11];

  STAGE_POINTS();
  __syncthreads();

  int half = lane >> 4, l = lane & 15;
  BUILD_OPS();
  float sxl[4], syl[4], szl[4];
#pragma unroll
  for (int i = 0; i < 4; ++i) {
    int idx = i * 16 + l;
    sxl[i] = sx[idx]; syl[i] = sy[idx]; szl[i] = sz[idx];
  }

  float S = 0, Srx = 0, Sry = 0, Srz = 0, Ssx = 0, Ssy = 0, Ssz = 0;
#pragma unroll
  for (int mi = 0; mi < 4; ++mi) {
    int mbase = mi * 16 + (half << 3);
    float r2loc[8], rxl[8], ryl[8], rzl[8];
#pragma unroll
    for (int j = 0; j < 8; ++j) {
      r2loc[j] = r2r[mbase + j];
      rxl[j] = rx[mbase + j]; ryl[j] = ry[mbase + j]; rzl[j] = rz[mbase + j];
    }
#pragma unroll
    for (int ni = 0; ni < 4; ++ni) {
      v8f acc = wmma_tile(aop[mi], bop[ni]);
      int n = ni * 16 + l;
#pragma unroll
      for (int j = 0; j < 8; ++j) {
        float wv = wb[(mbase + j) * NN + n];
        float d2 = r2loc[j] + s2loc[ni] - 2.0f * acc[j];
        float wp = (d2 < R2ACC) ? wv : 0.0f;     // branchless select
        S += wp;
        Srx += wp * rxl[j]; Sry += wp * ryl[j]; Srz += wp * rzl[j];
        Ssx += wp * sxl[ni]; Ssy += wp * syl[ni]; Ssz += wp * szl[ni];
      }
    }
  }
  red[lane][0] = S;
  red[lane][1] = Srx; red[lane][2] = Sry; red[lane][3] = Srz;
  red[lane][4] = Ssx; red[lane][5] = Ssy; red[lane][6] = Ssz;
  red[lane][7] = 0.0f;
  __syncthreads();
  for (int s = 16; s > 0; s >>= 1) {
    if (lane < s) for (int j = 0; j < 8; ++j) red[lane][j] += red[lane + s][j];
    __syncthreads();
  }
  if (lane == 0)
    for (int j = 0; j < 7; ++j) part1[b * 8 + j] = red[0][j];
}

// ---- K5b: reduce pass-1 partials -> cr, cs --------------------------------
__global__ void lgr_refine_reduce1(const float* __restrict__ part1,
                                   float* __restrict__ crcs) {
  int t = threadIdx.x;  // 32 threads
  __shared__ float s[8];
  if (t < 7) {
    float acc = 0.0f;
    for (int b = 0; b < BB; ++b) acc += part1[b * 8 + t];
    s[t] = acc;
  }
  __syncthreads();
  if (t == 0) {
    float Sp = s[0] + EPSV;
    crcs[0] = s[1] / Sp; crcs[1] = s[2] / Sp; crcs[2] = s[3] / Sp;
    crcs[3] = s[4] / Sp; crcs[4] = s[5] / Sp; crcs[5] = s[6] / Sp;
  }
}

// ---- K5c: refinement pass 2 (per-patch H partials via WMMA) ---------------
__global__ void lgr_refine_pass2(const float* __restrict__ ref,
                                 const float* __restrict__ src,
                                 const float* __restrict__ w,
                                 const float* __restrict__ curRT,
                                 const float* __restrict__ crcs,
                                 float* __restrict__ part2) {
  int b = blockIdx.x, lane = threadIdx.x;  // 32 threads
  __shared__ float rx[64], ry[64], rz[64], sx[64], sy[64], sz[64];
  __shared__ float mx[64], my[64], mz[64], r2r[64], s2m[64];
  __shared__ float red[32][12];
  const float* wb = w + b * MM * NN;

  float R00 = curRT[0], R01 = curRT[1], R02 = curRT[2];
  float R10 = curRT[3], R11 = curRT[4], R12 = curRT[5];
  float R20 = curRT[6], R21 = curRT[7], R22 = curRT[8];
  float t0 = curRT[9], t1 = curRT[10], t2 = curRT[11];
  float cr0 = crcs[0], cr1 = crcs[1], cr2 = crcs[2];
  float cs0 = crcs[3], cs1 = crcs[4], cs2 = crcs[5];

  STAGE_POINTS();
  __syncthreads();

  int half = lane >> 4, l = lane & 15;
  BUILD_OPS();
  float sc0l[4], sc1l[4], sc2l[4];
#pragma unroll
  for (int i = 0; i < 4; ++i) {
    int idx = i * 16 + l;
    sc0l[i] = sx[idx] - cs0; sc1l[i] = sy[idx] - cs1; sc2l[i] = sz[idx] - cs2;
  }

  float h[9] = {0, 0, 0, 0, 0, 0, 0, 0, 0};
#pragma unroll
  for (int mi = 0; mi < 4; ++mi) {
    int mbase = mi * 16 + (half << 3);
    float r2loc[8], rc0[8], rc1[8], rc2[8];
#pragma unroll
    for (int j = 0; j < 8; ++j) {
      r2loc[j] = r2r[mbase + j];
      rc0[j] = rx[mbase + j] - cr0;
      rc1[j] = ry[mbase + j] - cr1;
      rc2[j] = rz[mbase + j] - cr2;
    }
#pragma unroll
    for (int ni = 0; ni < 4; ++ni) {
      v8f acc = wmma_tile(aop[mi], bop[ni]);
      int n = ni * 16 + l;
#pragma unroll
      for (int j = 0; j < 8; ++j) {
        float wv = wb[(mbase + j) * NN + n];
        float d2 = r2loc[j] + s2loc[ni] - 2.0f * acc[j];
        float wp = (d2 < R2ACC) ? wv : 0.0f;
        h[0] += wp * sc0l[ni] * rc0[j];
        h[1] += wp * sc0l[ni] * rc1[j];
        h[2] += wp * sc0l[ni] * rc2[j];
        h[3] += wp * sc1l[ni] * rc0[j];
        h[4] += wp * sc1l[ni] * rc1[j];
        h[5] += wp * sc1l[ni] * rc2[j];
        h[6] += wp * sc2l[ni] * rc0[j];
        h[7] += wp * sc2l[ni] * rc1[j];
        h[8] += wp * sc2l[ni] * rc2[j];
      }
    }
  }
  for (int j = 0; j < 9; ++j) red[lane][j] = h[j];
  red[lane][9] = red[lane][10] = red[lane][11] = 0.0f;
  __syncthreads();
  for (int s = 16; s > 0; s >>= 1) {
    if (lane < s) for (int j = 0; j < 9; ++j) red[lane][j] += red[lane + s][j];
    __syncthreads();
  }
  if (lane == 0)
    for (int j = 0; j < 9; ++j) part2[b * 12 + j] = red[0][j];
}

// ---- K5d: reduce H partials, SVD, update curRT, write transform -----------
__global__ void lgr_refine_reduce2_svd(const float* __restrict__ part2,
                                       const float* __restrict__ crcs,
                                       float* __restrict__ curRT,
                                       float* __restrict__ outT) {
  int t = threadIdx.x;  // 32 threads
  __shared__ float h[12];
  if (t < 9) {
    float acc = 0.0f;
    for (int b = 0; b < BB; ++b) acc += part2[b * 12 + t];
    h[t] = acc;
  }
  __syncthreads();
  if (t == 0) {
    float H[9], cr[3] = {crcs[0], crcs[1], crcs[2]};
    float cs[3] = {crcs[3], crcs[4], crcs[5]};
    for (int j = 0; j < 9; ++j) H[j] = h[j];
    float R[9], tt[3];
    svd_rigid_rt(H, cr, cs, R, tt);
    for (int j = 0; j < 9; ++j) curRT[j] = R[j];
    curRT[9] = tt[0]; curRT[10] = tt[1]; curRT[11] = tt[2];
    outT[0] = R[0]; outT[1] = R[1]; outT[2]  = R[2]; outT[3]  = tt[0];
    outT[4] = R[3]; outT[5] = R[4]; outT[6]  = R[5]; outT[7]  = tt[1];
    outT[8] = R[6]; outT[9] = R[7]; outT[10] = R[8]; outT[11] = tt[2];
    outT[12] = 0.0f; outT[13] = 0.0f; outT[14] = 0.0f; outT[15] = 1.0f;
  }
}

// ---------------------------------------------------------------------------
extern "C" void kernel_launch(void* const* d_in, const int* in_sizes, int n_in,
                              void* d_out, int out_size, void* d_ws,
                              size_t ws_size, hipStream_t stream) {
  const float* ref = (const float*)d_in[0];    // (128,64,3)
  const float* src = (const float*)d_in[1];    // (128,64,3)
  const float* score = (const float*)d_in[2];  // (128,64,64)
  float* w = (float*)d_out;                    // output 1: (128,64,64)
  float* outT = (float*)d_out + BB * MM * NN;  // output 2: (4,4)

  // workspace carve (all 4-byte aligned)
  float* f = (float*)d_ws;
  float* Rb = f;      f += BB * 9;
  float* tb = f;      f += BB * 3;
  float* curRT = f;   f += 12;
  float* crcs = f;    f += 8;
  float* part1 = f;   f += BB * 8;
  float* part2 = f;   f += BB * 12;
  int* ip = (int*)f;
  int* corrcnt = ip;  ip += BB;
  int* pcount = ip;   ip += BB * BB;

  lgr_corr_kernel<<<BB, 64, 0, stream>>>(score, w, corrcnt);
  lgr_procrustes_batch_kernel<<<BB, 64, 0, stream>>>(ref, src, w, Rb, tb);
  lgr_verify_kernel<<<dim3(BB, BB), 32, 0, stream>>>(ref, src, w, Rb, tb,
                                                     pcount);
  lgr_select_best_kernel<<<1, BB, 0, stream>>>(pcount, corrcnt, Rb, tb, curRT);
  for (int it = 0; it < 5; ++it) {
    lgr_refine_pass1<<<BB, 32, 0, stream>>>(ref, src, w, curRT, part1);
    lgr_refine_reduce1<<<1, 32, 0, stream>>>(part1, crcs);
    lgr_refine_pass2<<<BB, 32, 0, stream>>>(ref, src, w, curRT, crcs, part2);
    lgr_refine_reduce2_svd<<<1, 32, 0, stream>>>(part2, crcs, curRT, outT);
  }
}